// FrequencyStripAttention_283467842728
// MI455X (gfx1250) — compile-verified
//
#include <hip/hip_runtime.h>
#include <stdint.h>

// ---------------------------------------------------------------------------
// FrequencyStripAttention for MI455X (gfx1250).
// HBM-bound separable 7-tap reflect box filter. Data path:
//   tensor_load_to_lds (TDM DMA, TENSORcnt)  ->  LDS
//   horizontal pass: float4 LDS loads, register sliding sums
//   vertical pass:   7-row register ring buffer, 1 LDS load per output row
//   coalesced global_store_b32 of each output row (no LDS round-trip)
// LDS = 77,824 B/WG -> 4 workgroups per 320 KB WGP for DMA/compute overlap.
// ---------------------------------------------------------------------------

typedef unsigned int v4u __attribute__((ext_vector_type(4)));
typedef int          v8i __attribute__((ext_vector_type(8)));
typedef int          v4i __attribute__((ext_vector_type(4)));

#define Wd      256
#define Hd      256
#define NCH     64
#define NB      16
#define TILE_H  32
#define MAXR    (TILE_H + 6)      // 38 rows incl. 3-row halo each side
#define TPB     256

#if __has_include(<hip/amd_detail/amd_gfx1250_TDM.h>)
#define TDM_SIX_ARGS 1
#else
#define TDM_SIX_ARGS 0
#endif

// 2-D tensor-DMA descriptor (D#): tile = rows x 256 fp32, row stride 256.
__device__ __forceinline__ void build_tdm_desc(unsigned long long gaddr,
                                               unsigned int ldsoff,
                                               int rows, v4u* g0, v8i* g1) {
  v4u a; v8i b;
  a[0] = 1u;                                                  // count=1, user mode
  a[1] = ldsoff;                                              // lds_addr (bytes)
  a[2] = (unsigned int)(gaddr & 0xFFFFFFFFull);               // global_addr[31:0]
  a[3] = (unsigned int)((gaddr >> 32) & 0x01FFFFFFull)        // global_addr[56:32]
       | (2u << 30);                                          // type = 2 ("image")
  b[0] = (int)(2u << 16);                                     // data_size = 4 bytes
  b[1] = (int)(256u << 16);                                   // tensor_dim0 = 256
  b[2] = (int)(((unsigned)rows) << 16);                       // tensor_dim1 = rows
  b[3] = (int)(256u << 16);                                   // tile_dim0 = 256
  b[4] = rows;                                                // tile_dim1 = rows
  b[5] = 256;                                                 // tensor_dim0_stride
  b[6] = 0;
  b[7] = 0;
  *g0 = a; *g1 = b;
}

__device__ __forceinline__ void tdm_load(v4u g0, v8i g1) {
  v4i z4 = {0, 0, 0, 0};
#if TDM_SIX_ARGS
  v8i z8 = {0, 0, 0, 0, 0, 0, 0, 0};
  __builtin_amdgcn_tensor_load_to_lds(g0, g1, z4, z4, z8, 0);
#else
  __builtin_amdgcn_tensor_load_to_lds(g0, g1, z4, z4, 0);
#endif
}

__device__ __forceinline__ int reflect256(int v) {
  v = (v < 0) ? -v : v;
  return (v > 255) ? (510 - v) : v;
}

__global__ void fsa_tdm_kernel(const float* __restrict__ x,
                               const float* __restrict__ vert_low,
                               const float* __restrict__ vert_high,
                               const float* __restrict__ hori_low,
                               const float* __restrict__ hori_high,
                               const float* __restrict__ gamma,
                               const float* __restrict__ beta,
                               float* __restrict__ out) {
  extern __shared__ float smem[];
  float* xs = smem;                    // MAXR x 256 : input tile (+halo)
  float* hs = smem + MAXR * Wd;        // MAXR x 256 : horizontal-pass result

  const int tid    = (int)threadIdx.x;
  const int tilesY = Hd / TILE_H;
  const int plane  = (int)blockIdx.x / tilesY;    // b*NCH + c
  const int tile   = (int)blockIdx.x % tilesY;
  const int c      = plane & (NCH - 1);
  const int y0     = tile * TILE_H;
  const int base   = (y0 - 3 > 0) ? (y0 - 3) : 0;
  const int topr   = (y0 + TILE_H + 3 < Hd) ? (y0 + TILE_H + 3) : Hd;
  const int nrows  = topr - base;

  const float* xplane = x   + (size_t)plane * (size_t)(Hd * Wd);
  float*       oplane = out + (size_t)plane * (size_t)(Hd * Wd);

  // ---- Stage 1: one TDM DMA of rows [base, topr) into LDS (wave 0 only) ---
  if (tid < 32) {
    v4u g0; v8i g1;
    unsigned long long ga = (unsigned long long)(uintptr_t)(xplane + (size_t)base * Wd);
    build_tdm_desc(ga, (unsigned int)(uintptr_t)xs, nrows, &g0, &g1);
    tdm_load(g0, g1);
    __builtin_amdgcn_s_wait_tensorcnt(0);
  }

  // Per-channel coefficients (loaded by all waves while the DMA flies):
  //   hori_out = a1*x + (b1/7)*sum7_w(x)
  //   out      = beta*x + g*a2*hori_out + (g*b2/7)*sum7_h(hori_out)
  const float a1  = hori_high[c] + 1.0f;
  const float b1  = (hori_low[c] - a1) * (1.0f / 7.0f);
  const float a2  = vert_high[c] + 1.0f;
  const float gm  = gamma[c];
  const float ga2 = gm * a2;
  const float gb2 = gm * (vert_low[c] - a2) * (1.0f / 7.0f);
  const float bt  = beta[c];

  __syncthreads();

  // ---- Stage 2: horizontal 7-tap, float4 per thread (b128 LDS traffic) ----
  // Chunk = 4 consecutive columns; window needs cols [4k-4, 4k+7] = 3 float4s.
  const int nchunk = nrows << 6;               // 64 chunks per row
  for (int ci = tid; ci < nchunk; ci += TPB) {
    const int row  = ci >> 6;
    const int ch   = ci & 63;
    const int col0 = ch << 2;
    const float* rp = xs + (row << 8);
    float c0, c1, c2, c3;                       // cols col0-4 .. col0-1
    if (ch > 0) {
      float4 L = *(const float4*)(rp + col0 - 4);
      c0 = L.x; c1 = L.y; c2 = L.z; c3 = L.w;
    } else {                                    // reflect: -4,-3,-2,-1 -> 4,3,2,1
      c0 = rp[4]; c1 = rp[3]; c2 = rp[2]; c3 = rp[1];
    }
    float4 M = *(const float4*)(rp + col0);     // cols col0 .. col0+3
    float c8, c9, c10;                          // cols col0+4 .. col0+6 (+7 unused)
    if (ch < 63) {
      float4 R = *(const float4*)(rp + col0 + 4);
      c8 = R.x; c9 = R.y; c10 = R.z;
    } else {                                    // reflect: 256,257,258 -> 254,253,252
      c8 = rp[254]; c9 = rp[253]; c10 = rp[252];
    }
    // Sliding 7-wide sums for the 4 outputs.
    float s0 = c1 + c2 + c3 + M.x + M.y + M.z + M.w;
    float s1 = s0 - c1 + c8;
    float s2 = s1 - c2 + c9;
    float s3 = s2 - c3 + c10;
    float4 hv;
    hv.x = a1 * M.x + b1 * s0;
    hv.y = a1 * M.y + b1 * s1;
    hv.z = a1 * M.z + b1 * s2;
    hv.w = a1 * M.w + b1 * s3;
    *(float4*)(hs + (row << 8) + col0) = hv;
  }
  __syncthreads();

  // ---- Stage 3: vertical 7-tap via register ring buffer, direct stores ----
  // One column per thread; one new LDS load per output row after priming.
  {
    const int xc = tid;
    float w0 = hs[((reflect256(y0 - 3) - base) << 8) + xc];
    float w1 = hs[((reflect256(y0 - 2) - base) << 8) + xc];
    float w2 = hs[((reflect256(y0 - 1) - base) << 8) + xc];
    float w3 = hs[((y0     - base) << 8) + xc];
    float w4 = hs[((y0 + 1 - base) << 8) + xc];
    float w5 = hs[((y0 + 2 - base) << 8) + xc];
    float w6 = hs[((y0 + 3 - base) << 8) + xc];
    float sum = w0 + w1 + w2 + w3 + w4 + w5 + w6;
#pragma unroll
    for (int yo = 0; yo < TILE_H; ++yo) {
      const int gy = y0 + yo;
      const float xv = xs[((gy - base) << 8) + xc];
      oplane[gy * Wd + xc] = bt * xv + ga2 * w3 + gb2 * sum;   // coalesced row
      if (yo < TILE_H - 1) {
        const float wn = hs[((reflect256(gy + 4) - base) << 8) + xc];
        sum += wn - w0;
        w0 = w1; w1 = w2; w2 = w3; w3 = w4; w4 = w5; w5 = w6; w6 = wn;
      }
    }
  }
}

extern "C" void kernel_launch(void* const* d_in, const int* in_sizes, int n_in,
                              void* d_out, int out_size, void* d_ws, size_t ws_size,
                              hipStream_t stream) {
  const float* x  = (const float*)d_in[0];
  const float* vl = (const float*)d_in[1];
  const float* vh = (const float*)d_in[2];
  const float* hl = (const float*)d_in[3];
  const float* hh = (const float*)d_in[4];
  const float* gm = (const float*)d_in[5];
  const float* bt = (const float*)d_in[6];
  float* out = (float*)d_out;

  constexpr int    tilesY = Hd / TILE_H;
  constexpr size_t smem   = (size_t)(2 * MAXR) * Wd * sizeof(float); // 77,824 B
  dim3 grid(NB * NCH * tilesY);   // 8192 workgroups
  fsa_tdm_kernel<<<grid, TPB, smem, stream>>>(x, vl, vh, hl, hh, gm, bt, out);
}